// SummarizationModel_42039139893314
// MI455X (gfx1250) — compile-verified
//
#include <hip/hip_runtime.h>

#define N_NODES 50000
#define H_DIM   64
#define OUT_DIM 32
#define N_RELS  16
#define N_BASES 8
#define N_EDGES 800000

typedef float v2f __attribute__((ext_vector_type(2)));
typedef float v8f __attribute__((ext_vector_type(8)));

// ---------------- helpers ----------------

__global__ void zero_f32(float* __restrict__ p, int n) {
    int i = blockIdx.x * blockDim.x + threadIdx.x;
    if (i < n) p[i] = 0.0f;
}

// W[r, p] = sum_b comp[r,b] * basis[b, p]   (p flattens in*out)
__global__ void combine_basis(const float* __restrict__ comp,
                              const float* __restrict__ basis,
                              float* __restrict__ W,
                              long perBasis, long total) {
    long i = (long)blockIdx.x * blockDim.x + threadIdx.x;
    if (i >= total) return;
    int  r = (int)(i / perBasis);
    long p = i - (long)r * perBasis;
    float acc = 0.0f;
#pragma unroll
    for (int b = 0; b < N_BASES; ++b)
        acc += comp[r * N_BASES + b] * basis[(long)b * perBasis + p];
    W[i] = acc;
}

// ---------------- fp32 WMMA GEMM ----------------
// C[r] = A @ B[r]; A:[M,64] row-major, B:[64,N] row-major (per relation r),
// C:[M,N] row-major (per relation r). blockDim = 256 (8 waves), each wave
// computes one 16-row band across all N columns.
//
// B is staged into LDS once per block in a *pair-swizzled* layout so each
// WMMA B-fragment is one ds_load_b64:
//   Bs[(pr*N + c)*2 + {0,1}] = { B[2*pr][c], B[2*pr+1][c] },  pr = 2*kk + half
// B fragments are register-blocked so the 16 WMMAs per n-tile issue
// back-to-back with only partial-count waits.
__launch_bounds__(256)
__global__ void gemm_f32_wmma(const float* __restrict__ A,
                              const float* __restrict__ B,
                              float* __restrict__ C,
                              int M, int N) {
    const int K = 64;
    __shared__ float Bs[64 * 64];             // up to 16 KB (N <= 64)

    int r = blockIdx.y;
    const float* Bp = B + (long)r * K * N;
    float*       Cp = C + (long)r * M * N;

    // Cooperative pair-swizzled stage of B into LDS.
    // Task t: pair-row pr (krow = 2*pr, 2*pr+1), column pair c, c+1.
    {
        const int nHalf = N >> 1;             // column pairs
        const int tasks = 32 * nHalf;         // 32 pair-rows (K/2)
        for (int t = threadIdx.x; t < tasks; t += 256) {
            int pr = t / nHalf;
            int c  = (t - pr * nHalf) * 2;
            v2f b0 = *(const v2f*)(Bp + (2 * pr    ) * N + c);  // B[2pr][c..c+1]
            v2f b1 = *(const v2f*)(Bp + (2 * pr + 1) * N + c);  // B[2pr+1][c..c+1]
            float4 w;                          // {Pair(pr,c), Pair(pr,c+1)}
            w.x = b0.x; w.y = b1.x; w.z = b0.y; w.w = b1.y;
            *(float4*)(Bs + ((pr * N + c) << 1)) = w;           // ds_store_b128
        }
    }
    __syncthreads();

    int lane = threadIdx.x & 31;
    int wave = threadIdx.x >> 5;
    int mtile = blockIdx.x * 8 + wave;
    if (mtile * 16 >= M) return;              // wave-uniform: EXEC stays all-ones

    int half = lane >> 4;                     // 0: lanes 0-15, 1: lanes 16-31
    int lq   = lane & 15;
    int m0   = mtile * 16;

    // Preload the 16 A fragments (K = 64 = 16 steps of 4).
    // A-frag lane layout: row m0+lq, cols {4k + 2*half, 4k + 2*half + 1}.
    v2f afrag[16];
    const float* arow = A + (long)(m0 + lq) * K;
#pragma unroll
    for (int kk = 0; kk < 16; ++kk)
        afrag[kk] = *(const v2f*)(arow + kk * 4 + 2 * half);

    for (int n0 = 0; n0 < N; n0 += 16) {
        // B-frag lane layout: rows {4k + 2*half, 4k + 2*half + 1}, col n0+lq.
        // One ds_load_b64 per fragment thanks to the pair swizzle.
        v2f bfrag[16];
#pragma unroll
        for (int kk = 0; kk < 16; ++kk) {
            int pr = kk * 2 + half;
            bfrag[kk] = *(const v2f*)(Bs + ((pr * N + n0 + lq) << 1));
        }
        v8f acc = {};
#pragma unroll
        for (int kk = 0; kk < 16; ++kk)
            acc = __builtin_amdgcn_wmma_f32_16x16x4_f32(
                false, afrag[kk], false, bfrag[kk], (short)0, acc, false, false);
        // C/D layout: VGPR v -> row m0 + v + 8*half, col n0 + lq
#pragma unroll
        for (int v = 0; v < 8; ++v)
            Cp[(long)(m0 + v + 8 * half) * N + n0 + lq] = acc[v];
    }
}

// ---------------- edge gather + atomic scatter ----------------
// agg[dst[e]] += T[etype[e], row(e), :] * norm[e]
// row(e) = feats[src[e]] (layer 1) or src[e] (feats == nullptr).
// One wave per edge; width is 64 (float2 per lane) or 32 (float per lane).
__global__ void edge_scatter(const float* __restrict__ T,
                             const int*  __restrict__ etype,
                             const int*  __restrict__ src,
                             const int*  __restrict__ dst,
                             const float* __restrict__ norm,
                             const int*  __restrict__ feats,   // nullable
                             float* __restrict__ agg,
                             int nE, int width) {
    int e = blockIdx.x * (blockDim.x >> 5) + (threadIdx.x >> 5);
    if (e >= nE) return;
    int lane = threadIdx.x & 31;
    int et = etype[e];
    int s  = src[e];
    int row = feats ? feats[s] : s;
    float nm = norm[e];
    const float* trow = T + ((long)et * N_NODES + row) * width;
    float*       arow = agg + (long)dst[e] * width;
    if (width == 64) {
        int j = lane * 2;
        v2f t = *(const v2f*)(trow + j);     // one global_load_b64 per lane
        atomicAdd(arow + j,     t.x * nm);
        atomicAdd(arow + j + 1, t.y * nm);
    } else {
        atomicAdd(arow + lane, trow[lane] * nm);
    }
}

// ---------------- epilogues ----------------

// h = relu(agg + loopw[feats[n]] + bias)   (layer 1)
__global__ void finalize_gather(const float* __restrict__ agg,
                                const float* __restrict__ loopw,
                                const int*  __restrict__ feats,
                                const float* __restrict__ bias,
                                float* __restrict__ h, int n, int width) {
    int i = blockIdx.x * blockDim.x + threadIdx.x;
    if (i >= n * width) return;
    int node = i / width, j = i - node * width;
    float v = agg[i] + loopw[(long)feats[node] * width + j] + bias[j];
    h[i] = v > 0.0f ? v : 0.0f;
}

// out = relu(agg + lt + bias)   (layers 2/3; width is a power of two)
__global__ void finalize_add(const float* __restrict__ agg,
                             const float* __restrict__ lt,
                             const float* __restrict__ bias,
                             float* __restrict__ out, int n, int width) {
    int i = blockIdx.x * blockDim.x + threadIdx.x;
    if (i >= n * width) return;
    float v = agg[i] + lt[i] + bias[i & (width - 1)];
    out[i] = v > 0.0f ? v : 0.0f;
}

// ---------------- launch ----------------

extern "C" void kernel_launch(void* const* d_in, const int* in_sizes, int n_in,
                              void* d_out, int out_size, void* d_ws, size_t ws_size,
                              hipStream_t stream) {
    const int*   feats  = (const int*)  d_in[0];
    const int*   etype  = (const int*)  d_in[1];
    const float* enorm  = (const float*)d_in[2];
    const int*   src    = (const int*)  d_in[3];
    const int*   dst    = (const int*)  d_in[4];
    const float* basis1 = (const float*)d_in[5];
    const float* comp1  = (const float*)d_in[6];
    const float* loop1  = (const float*)d_in[7];
    const float* bias1  = (const float*)d_in[8];
    const float* basis2 = (const float*)d_in[9];
    const float* comp2  = (const float*)d_in[10];
    const float* loop2  = (const float*)d_in[11];
    const float* bias2  = (const float*)d_in[12];
    const float* basis3 = (const float*)d_in[13];
    const float* comp3  = (const float*)d_in[14];
    const float* loop3  = (const float*)d_in[15];
    const float* bias3  = (const float*)d_in[16];

    // workspace layout (floats)
    float* big = (float*)d_ws;                               // 16*50000*64 (W1 / HW2 / HW3)
    float* h   = big + (size_t)N_RELS * N_NODES * H_DIM;     // 50000*64
    float* agg = h   + (size_t)N_NODES * H_DIM;              // 50000*64
    float* lt  = agg + (size_t)N_NODES * H_DIM;              // 50000*64
    float* W2  = lt  + (size_t)N_NODES * H_DIM;              // 16*64*64
    float* W3  = W2  + (size_t)N_RELS * H_DIM * H_DIM;       // 16*64*32

    const int NH = N_NODES * H_DIM;                          // 3,200,000
    const int NO = N_NODES * OUT_DIM;                        // 1,600,000
    const int edgeBlocks = (N_EDGES + 7) / 8;                // 8 waves/block, 1 edge/wave
    dim3 gemmGridR((N_NODES / 16 + 7) / 8, N_RELS);          // 391 x 16
    dim3 gemmGrid1((N_NODES / 16 + 7) / 8, 1);               // 391 x 1

    // ---- layer 1: one-hot input => W1 row gather ----
    {
        long perB = (long)N_NODES * H_DIM, total = (long)N_RELS * perB;
        combine_basis<<<(unsigned)((total + 255) / 256), 256, 0, stream>>>(comp1, basis1, big, perB, total);
        zero_f32<<<(NH + 255) / 256, 256, 0, stream>>>(agg, NH);
        edge_scatter<<<edgeBlocks, 256, 0, stream>>>(big, etype, src, dst, enorm, feats, agg, N_EDGES, H_DIM);
        finalize_gather<<<(NH + 255) / 256, 256, 0, stream>>>(agg, loop1, feats, bias1, h, N_NODES, H_DIM);
    }

    // ---- layer 2: h -> h ----
    {
        long perB = (long)H_DIM * H_DIM, total = (long)N_RELS * perB;
        combine_basis<<<(unsigned)((total + 255) / 256), 256, 0, stream>>>(comp2, basis2, W2, perB, total);
        gemm_f32_wmma<<<gemmGridR, 256, 0, stream>>>(h, W2, big, N_NODES, H_DIM);    // HW2[r] = h @ W2[r]
        gemm_f32_wmma<<<gemmGrid1, 256, 0, stream>>>(h, loop2, lt, N_NODES, H_DIM);  // self-loop term
        zero_f32<<<(NH + 255) / 256, 256, 0, stream>>>(agg, NH);
        edge_scatter<<<edgeBlocks, 256, 0, stream>>>(big, etype, src, dst, enorm, nullptr, agg, N_EDGES, H_DIM);
        finalize_add<<<(NH + 255) / 256, 256, 0, stream>>>(agg, lt, bias2, h, N_NODES, H_DIM);
    }

    // ---- layer 3: h -> out ----
    {
        long perB = (long)H_DIM * OUT_DIM, total = (long)N_RELS * perB;
        combine_basis<<<(unsigned)((total + 255) / 256), 256, 0, stream>>>(comp3, basis3, W3, perB, total);
        gemm_f32_wmma<<<gemmGridR, 256, 0, stream>>>(h, W3, big, N_NODES, OUT_DIM);   // HW3[r] = h @ W3[r]
        gemm_f32_wmma<<<gemmGrid1, 256, 0, stream>>>(h, loop3, lt, N_NODES, OUT_DIM);
        zero_f32<<<(NO + 255) / 256, 256, 0, stream>>>(agg, NO);
        edge_scatter<<<edgeBlocks, 256, 0, stream>>>(big, etype, src, dst, enorm, nullptr, agg, N_EDGES, OUT_DIM);
        finalize_add<<<(NO + 255) / 256, 256, 0, stream>>>(agg, lt, bias3, (float*)d_out, N_NODES, OUT_DIM);
    }
}